// GCN_34316788695393
// MI455X (gfx1250) — compile-verified
//
#include <hip/hip_runtime.h>

typedef __attribute__((ext_vector_type(2))) float v2f;
typedef __attribute__((ext_vector_type(8))) float v8f;

// ---------------------------------------------------------------------------
// Utility: fill a float buffer with a constant (used to zero accumulators and
// to init degrees to 1.0 for the self-loop contribution).
// ---------------------------------------------------------------------------
__global__ void gcn_fill_kernel(float* __restrict__ p, long long n, float v) {
    long long i = (long long)blockIdx.x * blockDim.x + threadIdx.x;
    if (i < n) p[i] = v;
}

// ---------------------------------------------------------------------------
// Degree accumulation: deg[dst[e]] += 1  (deg pre-initialized to 1.0 = self loop)
// ---------------------------------------------------------------------------
__global__ void gcn_deg_kernel(const int* __restrict__ dst, int E, float* __restrict__ deg) {
    int e = blockIdx.x * blockDim.x + threadIdx.x;
    if (e < E) atomicAdd(&deg[dst[e]], 1.0f);
}

__global__ void gcn_dinv_kernel(const float* __restrict__ deg, float* __restrict__ dinv, int n) {
    int i = blockIdx.x * blockDim.x + threadIdx.x;
    if (i < n) dinv[i] = rsqrtf(deg[i]);
}

// ---------------------------------------------------------------------------
// Full-precision fp32 WMMA GEMM:  C[M,Nout] = A[M,K] * W[K,Nout]
//   - one 16x16 C tile per wave, via V_WMMA_F32_16X16X4_F32 (K stepped by 4)
//   - A tile (16 x K) staged in LDS once, shared by all waves of the block
//   - blockDim.x = (Nout/16)*32, gridDim.x = ceil(M/16)
//   - K multiple of 4, Nout multiple of 16 (64 and 128 here)
// VGPR layouts per CDNA5 ISA 7.12.2:
//   A 16x4 f32 : lane%16 = M row; K = vgprIdx + 2*(lane/16)
//   B 4x16  f32: lane%16 = N col; K = vgprIdx + 2*(lane/16)
//   C/D 16x16  : lane%16 = N col; M = vgprIdx + 8*(lane/16)
// No divergent control flow: EXEC stays all-1s around every WMMA.
// ---------------------------------------------------------------------------
__global__ void gcn_gemm_wmma_kernel(const float* __restrict__ A,
                                     const float* __restrict__ W,
                                     float* __restrict__ C,
                                     int M, int K, int Nout) {
    extern __shared__ float ldsA[]; // 16 * K floats

    const int tid      = threadIdx.x;
    const int wave     = tid >> 5;
    const int lane     = tid & 31;
    const int laneHalf = lane >> 4;   // 0 or 1
    const int lane16   = lane & 15;
    const int mBase    = blockIdx.x * 16;

    // Cooperative stage of the 16 x K A-tile into LDS (clamp rows for M tail).
    const int tileElems = 16 * K;
    for (int i = tid; i < tileElems; i += blockDim.x) {
        int r = i / K;
        int c = i - r * K;
        int row = mBase + r;
        row = (row < M) ? row : (M - 1);
        ldsA[i] = A[(long long)row * K + c];
    }
    __syncthreads();

    const int n0 = wave * 16 + lane16; // output column for this lane

    v8f acc = {};
    for (int kk = 0; kk < K; kk += 4) {
        const int kb = kk + 2 * laneHalf; // K index for vgpr0 of the pair
        v2f a, b;
        a.x = ldsA[lane16 * K + kb];
        a.y = ldsA[lane16 * K + kb + 1];
        b.x = W[(long long)kb * Nout + n0];
        b.y = W[(long long)(kb + 1) * Nout + n0];
        // 8 args: (neg_a, A, neg_b, B, c_mod, C, reuse_a, reuse_b)
        acc = __builtin_amdgcn_wmma_f32_16x16x4_f32(
            /*neg_a=*/false, a, /*neg_b=*/false, b,
            /*c_mod=*/(short)0, acc, /*reuse_a=*/false, /*reuse_b=*/false);
    }

#pragma unroll
    for (int r = 0; r < 8; ++r) {
        int m = mBase + r + 8 * laneHalf;
        if (m < M) C[(long long)m * Nout + n0] = acc[r];
    }
}

// ---------------------------------------------------------------------------
// Edge scatter:  agg[dst] += dinv[src]*dinv[dst] * h[src]
//   LOGL = log2(lanes per edge), lanes-per-edge = D/4, each lane handles a
//   float4 slice (vector gather from L2-resident h, 4 fp32 global atomics).
// ---------------------------------------------------------------------------
template <int LOGL>
__global__ void gcn_scatter_kernel(const float* __restrict__ h,
                                   const int* __restrict__ src,
                                   const int* __restrict__ dst,
                                   const float* __restrict__ dinv,
                                   float* __restrict__ agg,
                                   int E, int D) {
    long long tid = (long long)blockIdx.x * blockDim.x + threadIdx.x;
    int e = (int)(tid >> LOGL);
    int l = (int)(tid & ((1 << LOGL) - 1));
    if (e < E) {
        int s = src[e];
        int d = dst[e];
        float norm = dinv[s] * dinv[d];
        const float4* hv = (const float4*)(h + (long long)s * D);
        float4 v = hv[l];
        float* o = agg + (long long)d * D + l * 4;
        atomicAdd(o + 0, v.x * norm);
        atomicAdd(o + 1, v.y * norm);
        atomicAdd(o + 2, v.z * norm);
        atomicAdd(o + 3, v.w * norm);
    }
}

// ---------------------------------------------------------------------------
// Finalize: out = (agg + dinv^2 * h + bias), optional ReLU.
// Safe to run in-place (out == agg): purely elementwise.
// ---------------------------------------------------------------------------
__global__ void gcn_finalize_kernel(const float* __restrict__ agg,
                                    const float* __restrict__ h,
                                    const float* __restrict__ dinv,
                                    const float* __restrict__ bias,
                                    float* __restrict__ out,
                                    int n, int D, int doRelu) {
    long long i = (long long)blockIdx.x * blockDim.x + threadIdx.x;
    if (i < (long long)n * D) {
        int node = (int)(i / D);
        int d    = (int)(i - (long long)node * D);
        float di = dinv[node];
        float v  = agg[i] + di * di * h[i] + bias[d];
        out[i] = doRelu ? fmaxf(v, 0.0f) : v;
    }
}

// ---------------------------------------------------------------------------
// Launch: 2-layer GCN forward.
// ---------------------------------------------------------------------------
extern "C" void kernel_launch(void* const* d_in, const int* in_sizes, int n_in,
                              void* d_out, int out_size, void* d_ws, size_t ws_size,
                              hipStream_t stream) {
    (void)n_in; (void)out_size; (void)ws_size;

    const float* x  = (const float*)d_in[0];
    const int*   ei = (const int*)d_in[1];
    const float* W1 = (const float*)d_in[2];
    const float* b1 = (const float*)d_in[3];
    const float* W2 = (const float*)d_in[4];
    const float* b2 = (const float*)d_in[5];

    const int hid  = in_sizes[3];            // 128
    const int outd = in_sizes[5];            // 64
    const int ind  = in_sizes[2] / hid;      // 64
    const int N    = in_sizes[0] / ind;      // 100000
    const int E    = in_sizes[1] / 2;        // 1600000

    const int* src = ei;
    const int* dst = ei + E;

    // Workspace layout (floats):
    //   h1   : N*hid   (later reused as h2 : N*outd)
    //   agg1 : N*hid   (later reused as z1, then agg2 : N*outd)
    //   deg  : N
    //   dinv : N
    float* h1   = (float*)d_ws;
    float* agg1 = h1 + (long long)N * hid;
    float* deg  = agg1 + (long long)N * hid;
    float* dinv = deg + N;
    float* h2   = h1;    // h1 dead after finalize-1
    float* agg2 = agg1;  // z1 dead after GEMM-2
    float* out  = (float*)d_out;

    const int T = 256;
    auto cdiv = [](long long a, long long b) { return (int)((a + b - 1) / b); };

    // ---- normalization: deg (with self loop) -> dinv -------------------------
    gcn_fill_kernel<<<cdiv(N, T), T, 0, stream>>>(deg, N, 1.0f);
    gcn_deg_kernel<<<cdiv(E, T), T, 0, stream>>>(dst, E, deg);
    gcn_dinv_kernel<<<cdiv(N, T), T, 0, stream>>>(deg, dinv, N);

    // ---- layer 1: h1 = x @ W1 (fp32 WMMA) -----------------------------------
    {
        int grid  = cdiv(N, 16);
        int block = (hid / 16) * 32;          // 256 threads = 8 waves
        int shmem = 16 * ind * sizeof(float); // 4 KB
        gcn_gemm_wmma_kernel<<<grid, block, shmem, stream>>>(x, W1, h1, N, ind, hid);
    }
    gcn_fill_kernel<<<cdiv((long long)N * hid, T), T, 0, stream>>>(agg1, (long long)N * hid, 0.0f);
    gcn_scatter_kernel<5><<<cdiv((long long)E * 32, T), T, 0, stream>>>(h1, src, dst, dinv, agg1, E, hid);
    gcn_finalize_kernel<<<cdiv((long long)N * hid, T), T, 0, stream>>>(agg1, h1, dinv, b1, agg1, N, hid, 1);

    // ---- layer 2: h2 = z1 @ W2 (fp32 WMMA) ----------------------------------
    {
        int grid  = cdiv(N, 16);
        int block = (outd / 16) * 32;         // 128 threads = 4 waves
        int shmem = 16 * hid * sizeof(float); // 8 KB
        gcn_gemm_wmma_kernel<<<grid, block, shmem, stream>>>(agg1, W2, h2, N, hid, outd);
    }
    gcn_fill_kernel<<<cdiv((long long)N * outd, T), T, 0, stream>>>(agg2, (long long)N * outd, 0.0f);
    gcn_scatter_kernel<4><<<cdiv((long long)E * 16, T), T, 0, stream>>>(h2, src, dst, dinv, agg2, E, outd);
    gcn_finalize_kernel<<<cdiv((long long)N * outd, T), T, 0, stream>>>(agg2, h2, dinv, b2, out, N, outd, 0);
}